// GAT_GNN_21363167330510
// MI455X (gfx1250) — compile-verified
//
#include <hip/hip_runtime.h>
#include <math.h>

typedef __attribute__((ext_vector_type(2))) float v2f;
typedef __attribute__((ext_vector_type(8))) float v8f;

#define NEG_SLOPE 0.2f

// ---------------------------------------------------------------------------
// utility fills
// ---------------------------------------------------------------------------
__global__ void gat_fill_f32(float* p, float v, int n) {
    int i = blockIdx.x * blockDim.x + threadIdx.x;
    if (i < n) p[i] = v;
}

// ---------------------------------------------------------------------------
// fp32 WMMA GEMM: C[M,N] = A[M,K] @ B[K,N], row-major, N = NT*16.
// One wave computes one 16-row tile of A against ALL NT column tiles,
// so A is streamed from HBM exactly once (51 MB -> ~2us at 23.3 TB/s).
// Uses V_WMMA_F32_16X16X4_F32 (exact fp32; this GEMM is bandwidth-bound,
// so lower-precision WMMA would buy nothing).
//
// Per ISA layouts (CDNA5 ISA):
//   A 16x4 : lane 0-15 -> M=lane,K={k,k+1};  lane 16-31 -> M=lane-16,K={k+2,k+3}
//   B 4x16 : vgpr v, lanes 0-15 -> row k+2*half+v? (half=lane>>4): col = lane&15
//   C/D    : vgpr r, M = r + 8*(lane>>4), N = lane&15
// ---------------------------------------------------------------------------
template <int NT>
__global__ void gat_gemm_wmma_f32(const float* __restrict__ A,
                                  const float* __restrict__ B,
                                  float* __restrict__ C, int M, int K) {
    const int Nout  = NT * 16;
    const int wavesPerBlock = blockDim.x >> 5;
    const int mt    = blockIdx.x * wavesPerBlock + (threadIdx.x >> 5);
    const int lane  = threadIdx.x & 31;
    const int half  = lane >> 4;      // which K-pair this lane holds
    const int row   = lane & 15;      // M row (A) / N col (B, C)

    if (mt * 16 >= M) return;

    v8f acc[NT];
#pragma unroll
    for (int t = 0; t < NT; ++t) acc[t] = {};

    const float* __restrict__ arow = A + (size_t)(mt * 16 + row) * K;

    for (int k = 0; k < K; k += 4) {
        // A fragment: two consecutive fp32 at K = k + 2*half  (8B aligned)
        v2f a = *(const v2f*)(arow + k + 2 * half);
#pragma unroll
        for (int t = 0; t < NT; ++t) {
            v2f b;
            b.x = B[(size_t)(k + 2 * half) * Nout + t * 16 + row];
            b.y = B[(size_t)(k + 2 * half + 1) * Nout + t * 16 + row];
            acc[t] = __builtin_amdgcn_wmma_f32_16x16x4_f32(
                false, a, false, b, (short)0, acc[t], false, false);
        }
    }

#pragma unroll
    for (int t = 0; t < NT; ++t) {
#pragma unroll
        for (int r = 0; r < 8; ++r) {
            C[(size_t)(mt * 16 + r + 8 * half) * Nout + t * 16 + row] = acc[t][r];
        }
    }
}

// ---------------------------------------------------------------------------
// alpha_src[n,h] = sum_c xw[n,h,c]*a_src[h,c]  (C == 32 == wave32 lanes)
// one wave per (n,h); shuffle-reduce across the wave.
// ---------------------------------------------------------------------------
__global__ void gat_alpha(const float* __restrict__ xw,
                          const float* __restrict__ a_src,
                          const float* __restrict__ a_dst,
                          float* __restrict__ as_out, float* __restrict__ ad_out,
                          int N, int H) {
    const int C = 32;
    int w    = blockIdx.x * (blockDim.x >> 5) + (threadIdx.x >> 5);
    int lane = threadIdx.x & 31;
    if (w >= N * H) return;
    int h = w % H;
    float v  = xw[(size_t)w * C + lane];
    float s1 = v * a_src[h * C + lane];
    float s2 = v * a_dst[h * C + lane];
#pragma unroll
    for (int off = 16; off > 0; off >>= 1) {
        s1 += __shfl_xor(s1, off);
        s2 += __shfl_xor(s2, off);
    }
    if (lane == 0) { as_out[w] = s1; ad_out[w] = s2; }
}

// ---------------------------------------------------------------------------
// float atomic max via int CAS (handles negatives; init is -inf)
// ---------------------------------------------------------------------------
__device__ __forceinline__ void atomicMaxF32(float* addr, float val) {
    int* ai  = (int*)addr;
    int  old = __float_as_int(*addr);
    while (__int_as_float(old) < val) {
        int prev = atomicCAS(ai, old, __float_as_int(val));
        if (prev == old) break;
        old = prev;
    }
}

__device__ __forceinline__ float leaky(float x) {
    return x >= 0.f ? x : NEG_SLOPE * x;
}

// segment max of leaky-relu'd logits, one thread per edge
__global__ void gat_edge_max(const float* __restrict__ as,
                             const float* __restrict__ ad,
                             const int* __restrict__ src,
                             const int* __restrict__ dst,
                             float* __restrict__ m, int E, int H) {
    int e = blockIdx.x * blockDim.x + threadIdx.x;
    if (e >= E) return;
    int sj = src[e], di = dst[e];
    for (int h = 0; h < H; ++h) {
        float l = leaky(as[sj * H + h] + ad[di * H + h]);
        atomicMaxF32(&m[di * H + h], l);
    }
}

// segment sum of exp(logit - max), one thread per edge
__global__ void gat_edge_sum(const float* __restrict__ as,
                             const float* __restrict__ ad,
                             const int* __restrict__ src,
                             const int* __restrict__ dst,
                             const float* __restrict__ m,
                             float* __restrict__ s, int E, int H) {
    int e = blockIdx.x * blockDim.x + threadIdx.x;
    if (e >= E) return;
    int sj = src[e], di = dst[e];
    for (int h = 0; h < H; ++h) {
        float l = leaky(as[sj * H + h] + ad[di * H + h]);
        atomicAdd(&s[di * H + h], expf(l - m[di * H + h]));
    }
}

// weighted scatter-add: one wave per edge, lane = channel (C==32).
// attention weight recomputed per edge (cheaper than storing [E,H] and
// re-reading it: the 4 scalar gathers are wave-uniform -> broadcast loads).
__global__ void gat_edge_scatter(const float* __restrict__ as,
                                 const float* __restrict__ ad,
                                 const float* __restrict__ m,
                                 const float* __restrict__ ssum,
                                 const float* __restrict__ xw,
                                 const int* __restrict__ src,
                                 const int* __restrict__ dst,
                                 float* __restrict__ agg, int E, int H) {
    const int C = 32;
    int e    = blockIdx.x * (blockDim.x >> 5) + (threadIdx.x >> 5);
    int lane = threadIdx.x & 31;
    if (e >= E) return;
    int sj = src[e], di = dst[e];
    for (int h = 0; h < H; ++h) {
        float l = leaky(as[sj * H + h] + ad[di * H + h]);
        float w = expf(l - m[di * H + h]) / (ssum[di * H + h] + 1e-16f);
        atomicAdd(&agg[(size_t)(di * H + h) * C + lane],
                  w * xw[(size_t)(sj * H + h) * C + lane]);
    }
}

// out[n,f] = (agg[n,f] + b[f]) with optional relu; in-place safe
__global__ void gat_finalize(const float* __restrict__ agg,
                             const float* __restrict__ b,
                             float* __restrict__ out, int total, int F,
                             int doRelu) {
    int i = blockIdx.x * blockDim.x + threadIdx.x;
    if (i >= total) return;
    float v = agg[i] + b[i % F];
    if (doRelu) v = fmaxf(v, 0.f);
    out[i] = v;
}

// ---------------------------------------------------------------------------
// launch
// ---------------------------------------------------------------------------
static inline int ceil_div(long long a, int b) { return (int)((a + b - 1) / b); }

extern "C" void kernel_launch(void* const* d_in, const int* in_sizes, int n_in,
                              void* d_out, int out_size, void* d_ws, size_t ws_size,
                              hipStream_t stream) {
    const float* x      = (const float*)d_in[0];
    const int*   eidx   = (const int*)d_in[1];
    const float* W1     = (const float*)d_in[2];
    const float* a_src1 = (const float*)d_in[3];
    const float* a_dst1 = (const float*)d_in[4];
    const float* b1     = (const float*)d_in[5];
    const float* W2     = (const float*)d_in[6];
    const float* a_src2 = (const float*)d_in[7];
    const float* a_dst2 = (const float*)d_in[8];
    const float* b2     = (const float*)d_in[9];

    const int F_in = 128, F1 = 64, F2 = 32, H1 = 2, H2 = 1;
    const int N = in_sizes[0] / F_in;
    const int E = in_sizes[1] / 2;
    const int* src = eidx;        // edge_index[0]
    const int* dst = eidx + E;    // edge_index[1]

    // workspace layout (floats)
    float* ws   = (float*)d_ws;
    float* xw1  = ws;                       // N*64
    float* h1   = xw1 + (size_t)N * F1;     // N*64 (agg1, then relu'd features)
    float* as1  = h1  + (size_t)N * F1;     // N*H1
    float* ad1  = as1 + (size_t)N * H1;
    float* m1   = ad1 + (size_t)N * H1;
    float* s1   = m1  + (size_t)N * H1;
    float* xw2  = s1  + (size_t)N * H1;     // N*32
    float* as2  = xw2 + (size_t)N * F2;     // N
    float* ad2  = as2 + (size_t)N;
    float* m2   = ad2 + (size_t)N;
    float* s2   = m2  + (size_t)N;
    float* agg2 = s2  + (size_t)N;          // N*32

    const float NEG_INF = -__builtin_inff();
    const int TB = 256;

    // ---------------- layer 1 ----------------
    // xw1 = x @ W1   (fp32 WMMA, NT=4 -> Nout=64)
    {
        int waves = ceil_div(N, 16);
        gat_gemm_wmma_f32<4><<<ceil_div((long long)waves * 32, TB), TB, 0, stream>>>(
            x, W1, xw1, N, F_in);
    }
    gat_alpha<<<ceil_div((long long)N * H1 * 32, TB), TB, 0, stream>>>(
        xw1, a_src1, a_dst1, as1, ad1, N, H1);

    gat_fill_f32<<<ceil_div((long long)N * F1, TB), TB, 0, stream>>>(h1, 0.f, N * F1);
    gat_fill_f32<<<ceil_div((long long)N * H1, TB), TB, 0, stream>>>(m1, NEG_INF, N * H1);
    gat_fill_f32<<<ceil_div((long long)N * H1, TB), TB, 0, stream>>>(s1, 0.f, N * H1);

    gat_edge_max<<<ceil_div(E, TB), TB, 0, stream>>>(as1, ad1, src, dst, m1, E, H1);
    gat_edge_sum<<<ceil_div(E, TB), TB, 0, stream>>>(as1, ad1, src, dst, m1, s1, E, H1);
    gat_edge_scatter<<<ceil_div((long long)E * 32, TB), TB, 0, stream>>>(
        as1, ad1, m1, s1, xw1, src, dst, h1, E, H1);
    gat_finalize<<<ceil_div((long long)N * F1, TB), TB, 0, stream>>>(
        h1, b1, h1, N * F1, F1, /*relu=*/1);

    // ---------------- layer 2 ----------------
    // xw2 = h1 @ W2  (fp32 WMMA, NT=2 -> Nout=32)
    {
        int waves = ceil_div(N, 16);
        gat_gemm_wmma_f32<2><<<ceil_div((long long)waves * 32, TB), TB, 0, stream>>>(
            h1, W2, xw2, N, F1);
    }
    gat_alpha<<<ceil_div((long long)N * H2 * 32, TB), TB, 0, stream>>>(
        xw2, a_src2, a_dst2, as2, ad2, N, H2);

    gat_fill_f32<<<ceil_div((long long)N * F2, TB), TB, 0, stream>>>(agg2, 0.f, N * F2);
    gat_fill_f32<<<ceil_div(N, TB), TB, 0, stream>>>(m2, NEG_INF, N);
    gat_fill_f32<<<ceil_div(N, TB), TB, 0, stream>>>(s2, 0.f, N);

    gat_edge_max<<<ceil_div(E, TB), TB, 0, stream>>>(as2, ad2, src, dst, m2, E, H2);
    gat_edge_sum<<<ceil_div(E, TB), TB, 0, stream>>>(as2, ad2, src, dst, m2, s2, E, H2);
    gat_edge_scatter<<<ceil_div((long long)E * 32, TB), TB, 0, stream>>>(
        as2, ad2, m2, s2, xw2, src, dst, agg2, E, H2);
    gat_finalize<<<ceil_div((long long)N * F2, TB), TB, 0, stream>>>(
        agg2, b2, (float*)d_out, N * F2, F2, /*relu=*/0);
}